// Model_50431505990111
// MI455X (gfx1250) — compile-verified
//
#include <hip/hip_runtime.h>
#include <math.h>

typedef __attribute__((ext_vector_type(2))) float v2f;
typedef __attribute__((ext_vector_type(8))) float v8f;
typedef __attribute__((ext_vector_type(4))) unsigned int v4u;
typedef __attribute__((ext_vector_type(4))) int v4i;
typedef __attribute__((ext_vector_type(8))) int v8i;

#define ST 30
#define SX 64
#define GRID_N 1920
#define T_FULL 60
#define X_FULL 128
#define NPIX 7680
#define MU_ 10000.0f
#define LAM_ 1e-6f
#define HT 0.05f
#define HX 0.03f

#if __has_builtin(__builtin_amdgcn_tensor_load_to_lds)
#define USE_TDM 1
#endif

__device__ static inline unsigned lds_off(const void* p) {
    // low 32 bits of a flat shared-space address = LDS byte offset (ISA §10.2)
    return (unsigned)(unsigned long long)(uintptr_t)p;
}

#ifdef USE_TDM
// Issue one TDM 2D tile load: 32x32 f32 tile, row stride GRID_N, into LDS.
// D# layout per CDNA5 ISA §8.3/§8.4 (group0: count/lds_addr/global_addr/type=2,
// group1: data_size=4B, dims/strides/tile dims). Groups 2/3 zero (<=2D tensor).
__device__ static inline void tdm_load_tile32(const float* gsrc, unsigned lds_byte_off) {
    unsigned long long ga = (unsigned long long)(uintptr_t)gsrc;
    v4u g0;
    g0[0] = 1u;                                   // count=1, no gather
    g0[1] = lds_byte_off;                         // lds_addr
    g0[2] = (unsigned)(ga & 0xFFFFFFFFu);         // global_addr[31:0]
    g0[3] = (unsigned)(ga >> 32) | (2u << 30);    // global_addr[56:32] | type=2
    v8i g1;
    g1[0] = (int)(2u << 16);                      // data_size=2 (4 bytes)
    g1[1] = (int)(1920u << 16);                   // tensor_dim0[15:0] in bits[63:48]
    g1[2] = (int)(1920u << 16);                   // dim0 hi=0 | tensor_dim1[15:0]
    g1[3] = (int)(32u << 16);                     // dim1 hi=0 | tile_dim0=32
    g1[4] = 32;                                   // tile_dim1=32, tile_dim2=0
    g1[5] = 1920;                                 // tensor_dim0_stride[31:0]
    g1[6] = 0;                                    // stride hi | dim1_stride (unused 2D)
    g1[7] = 0;
    v4i z4 = {0, 0, 0, 0};
#if __clang_major__ >= 23
    v8i z8 = {0, 0, 0, 0, 0, 0, 0, 0};
    __builtin_amdgcn_tensor_load_to_lds(g0, g1, z4, z4, z8, 0);
#else
    __builtin_amdgcn_tensor_load_to_lds(g0, g1, z4, z4, 0);
#endif
}
__device__ static inline void tdm_wait() {
#if __has_builtin(__builtin_amdgcn_s_wait_tensorcnt)
    __builtin_amdgcn_s_wait_tensorcnt(0);
#else
    asm volatile("s_wait_tensorcnt 0x0" ::: "memory");
#endif
}
#endif

// ---------------------------------------------------------------- conv stack
__global__ void conv_first(const float* __restrict__ u, const float* __restrict__ w,
                           const float* __restrict__ b, float* __restrict__ out) {
    int idx = blockIdx.x * blockDim.x + threadIdx.x;      // co*NPIX + pixel
    if (idx >= 64 * NPIX) return;
    int co = idx / NPIX, p = idx % NPIX;
    int y = p / X_FULL, x = p % X_FULL;
    float acc = b[co];
    for (int ky = 0; ky < 5; ky++) {
        int iy = y + ky - 2;
        for (int kx = 0; kx < 5; kx++) {
            int ix = x + kx - 2;
            bool ok = (unsigned)iy < (unsigned)T_FULL && (unsigned)ix < (unsigned)X_FULL;
            int iyc = min(max(iy, 0), T_FULL - 1), ixc = min(max(ix, 0), X_FULL - 1);
            float v = u[iyc * X_FULL + ixc];
            acc += (ok ? v : 0.0f) * w[co * 25 + ky * 5 + kx];
        }
    }
    out[co * NPIX + p] = fmaxf(acc, 0.0f);                // ReLU (layer 0 < 6)
}

// Pre-transpose a 64->64 conv weight tensor (co,ci,5,5) -> wT[tap][ci][co]
// so the WMMA B-operand stream is lane-coalesced and tap-local (no spills).
__global__ void transpose_w(const float* __restrict__ w, float* __restrict__ wT) {
    int idx = blockIdx.x * blockDim.x + threadIdx.x;      // tap*4096 + ci*64 + co
    if (idx >= 25 * 4096) return;
    int tap = idx >> 12, r = idx & 4095, ci = r >> 6, co = r & 63;
    wT[idx] = w[(size_t)co * 1600 + ci * 25 + tap];
}

// 64->64 layer: implicit GEMM with V_WMMA_F32_16X16X4_F32.
// One block = 128 threads (4 waves). Block handles 16 pixels x 64 out-channels;
// wave w owns co tile [16w, 16w+16). Input halo tile (64ci x 5x20) staged in LDS
// with boundary zeros resolved at fill time -> branch-free WMMA loop.
// LDS layout [pos][ci], row stride 66 floats (8B-aligned v2f, bank-spread).
// Weights come from the pre-transposed wT[tap][ci][co] (lane-coalesced).
__global__ void conv_mid_wmma(const float* __restrict__ in, const float* __restrict__ wT,
                              const float* __restrict__ bias, float* __restrict__ out) {
    __shared__ float Tin[100 * 66];
    int tid = threadIdx.x;
    int p0 = blockIdx.x * 16;
    int y0 = p0 / X_FULL, x0 = p0 % X_FULL;               // 16-pixel tile is one row
    for (int idx = tid; idx < 6400; idx += 128) {
        int ci = idx / 100, pos = idx % 100;              // consecutive tid -> coalesced x
        int dy = pos / 20, dxp = pos % 20;
        int iy = y0 + dy - 2, ix = x0 + dxp - 2;
        bool ok = (unsigned)iy < (unsigned)T_FULL && (unsigned)ix < (unsigned)X_FULL;
        int iyc = min(max(iy, 0), T_FULL - 1), ixc = min(max(ix, 0), X_FULL - 1);
        float v = in[(size_t)ci * NPIX + iyc * X_FULL + ixc];
        Tin[pos * 66 + ci] = ok ? v : 0.0f;
    }
    __syncthreads();

    int wv = tid >> 5, lane = tid & 31;
    int co0 = wv * 16;
    int mcol = lane & 15;        // M row (pixel) for A, N col (co) for B/C
    int khalf = lane >> 4;       // K base selector
    int cocol = co0 + mcol;
    float bval = bias[cocol];
    v8f c;
    for (int r = 0; r < 8; r++) c[r] = bval;
#pragma unroll 1
    for (int ky = 0; ky < 5; ky++) {
#pragma unroll 1
        for (int kx = 0; kx < 5; kx++) {
            int tap = ky * 5 + kx;
            int pos = ky * 20 + mcol + kx;                // A-operand pixel offset
            const float* wt = wT + (size_t)tap * 4096;
            __builtin_prefetch(wt + 4096 + cocol, 0, 2);  // next tap's weights
#pragma unroll 4
            for (int k0 = 0; k0 < 64; k0 += 4) {
                int ci = k0 + 2 * khalf;
                v2f a = *(const v2f*)&Tin[pos * 66 + ci]; // ds_load_b64
                v2f bm;
                bm.x = wt[(ci    ) * 64 + cocol];         // coalesced across lanes
                bm.y = wt[(ci + 1) * 64 + cocol];
                c = __builtin_amdgcn_wmma_f32_16x16x4_f32(false, a, false, bm,
                                                          (short)0, c, false, false);
            }
        }
    }
    for (int r = 0; r < 8; r++) {
        int prow = p0 + r + 8 * khalf;                    // C: VGPR r -> rows r / r+8
        out[(size_t)cocol * NPIX + prow] = fmaxf(c[r], 0.0f);
    }
}

__global__ void conv_last(const float* __restrict__ in, const float* __restrict__ w,
                          const float* __restrict__ b, float* __restrict__ out) {
    int p = blockIdx.x * blockDim.x + threadIdx.x;
    if (p >= NPIX) return;
    int y = p / X_FULL, x = p % X_FULL;
    float acc = b[0];
    for (int ci = 0; ci < 64; ci++)
        for (int ky = 0; ky < 5; ky++) {
            int iy = y + ky - 2;
            for (int kx = 0; kx < 5; kx++) {
                int ix = x + kx - 2;
                bool ok = (unsigned)iy < (unsigned)T_FULL && (unsigned)ix < (unsigned)X_FULL;
                int iyc = min(max(iy, 0), T_FULL - 1), ixc = min(max(ix, 0), X_FULL - 1);
                float v = in[(size_t)ci * NPIX + iyc * X_FULL + ixc];
                acc += (ok ? v : 0.0f) * w[ci * 25 + ky * 5 + kx];
            }
        }
    out[p] = acc;                                         // no ReLU on last layer
}

// ------------------------------------------------- dense A from kron stencils
__global__ void build_A(float* __restrict__ A, const float* __restrict__ coef) {
    int g = blockIdx.x;
    float* row = A + (size_t)g * GRID_N;
    for (int h = threadIdx.x; h < GRID_N; h += blockDim.x) row[h] = 0.0f;
    __syncthreads();
    if (threadIdx.x == 0) {
        float c0 = coef[0], c1 = coef[1], c2 = coef[2], c3 = coef[3], c4 = coef[4];
        int tt = g >> 6, xx = g & 63;
        row[g] += c0;                                     // identity
        if (tt == 0)          { row[0 * SX + xx] += -c1 / HT; row[1 * SX + xx] += c1 / HT; }
        else if (tt == ST - 1){ row[(ST-2)*SX+xx] += -c1 / HT; row[(ST-1)*SX+xx] += c1 / HT; }
        else { row[(tt-1)*SX+xx] += -0.5f * c1 / HT; row[(tt+1)*SX+xx] += 0.5f * c1 / HT; }
        if (xx == 0)          { row[tt*SX + 0] += -c2 / HX; row[tt*SX + 1] += c2 / HX; }
        else if (xx == SX - 1){ row[tt*SX+SX-2] += -c2 / HX; row[tt*SX+SX-1] += c2 / HX; }
        else { row[tt*SX+xx-1] += -0.5f * c2 / HX; row[tt*SX+xx+1] += 0.5f * c2 / HX; }
        float it2 = c3 / (HT * HT);
        int t0 = (tt == 0) ? 0 : (tt == ST - 1 ? ST - 3 : tt - 1);
        row[t0*SX+xx] += it2; row[(t0+1)*SX+xx] += -2.0f * it2; row[(t0+2)*SX+xx] += it2;
        float ix2 = c4 / (HX * HX);
        int x0 = (xx == 0) ? 0 : (xx == SX - 1 ? SX - 3 : xx - 1);
        row[tt*SX+x0] += ix2; row[tt*SX+x0+1] += -2.0f * ix2; row[tt*SX+x0+2] += ix2;
    }
}

// --------------------------- M = A^T A (WMMA f32, TDM-staged 32x32 K-panels)
__global__ void gemm_ata(const float* __restrict__ A, float* __restrict__ M) {
    __shared__ float As[32 * 32];
    __shared__ float Bs[32 * 32];
    int tid = threadIdx.x, wv = tid >> 5, lane = tid & 31;
    int h0 = blockIdx.x * 32, l0 = blockIdx.y * 32;
    int m = lane & 15, kh = lane >> 4;
    int si = wv >> 1, sj = wv & 1;
    v8f c = {};
#pragma unroll 1
    for (int g0i = 0; g0i < GRID_N; g0i += 32) {
#ifdef USE_TDM
        if (wv == 0) {
            tdm_load_tile32(A + (size_t)g0i * GRID_N + h0, lds_off(As));
            tdm_load_tile32(A + (size_t)g0i * GRID_N + l0, lds_off(Bs));
            tdm_wait();
        }
        __syncthreads();
#else
        for (int idx = tid; idx < 1024; idx += 128) {
            int r = idx >> 5, cc = idx & 31;
            As[idx] = A[(size_t)(g0i + r) * GRID_N + h0 + cc];
            Bs[idx] = A[(size_t)(g0i + r) * GRID_N + l0 + cc];
        }
        __syncthreads();
#endif
        for (int kk = 0; kk < 32; kk += 4) {
            int k = kk + 2 * kh;
            v2f a, b;
            a.x = As[k * 32 + si * 16 + m];
            a.y = As[(k + 1) * 32 + si * 16 + m];
            b.x = Bs[k * 32 + sj * 16 + m];
            b.y = Bs[(k + 1) * 32 + sj * 16 + m];
            c = __builtin_amdgcn_wmma_f32_16x16x4_f32(false, a, false, b,
                                                      (short)0, c, false, false);
        }
        __syncthreads();
    }
    for (int r = 0; r < 8; r++) {
        int h = h0 + si * 16 + r + 8 * kh;
        M[(size_t)h * GRID_N + l0 + sj * 16 + m] = c[r];
    }
}

__global__ void add_reg(float* __restrict__ M) {
    int h = blockIdx.x * blockDim.x + threadIdx.x;
    if (h >= GRID_N) return;
    int tt = h >> 6, xx = h & 63;
    float bm = (tt == 0 || tt == ST - 1 || xx == 0 || xx == SX - 1) ? 1.0f : 0.0f;
    M[(size_t)h * GRID_N + h] += MU_ * bm + LAM_;
}

// --------------------------------------------------------------- patch gather
__global__ void gather_patches(const float* __restrict__ u, const float* __restrict__ rhsf,
                               float* __restrict__ u_ch, float* __restrict__ rhs_p) {
    int i = blockIdx.x * blockDim.x + threadIdx.x;
    if (i >= 4 * GRID_N) return;
    int bp = i / GRID_N, g = i % GRID_N;
    int pt = bp >> 1, px = bp & 1;
    int tt = g >> 6, xx = g & 63;
    int idx = (pt * ST + tt) * X_FULL + px * SX + xx;
    u_ch[i]  = u[idx];
    rhs_p[i] = rhsf[idx];
}

__global__ void build_rhsv(const float* __restrict__ A, const float* __restrict__ rhs_p,
                           const float* __restrict__ u_ch, float* __restrict__ rhsv) {
    int i = blockIdx.x * blockDim.x + threadIdx.x;
    if (i >= 4 * GRID_N) return;
    int bp = i / GRID_N, h = i % GRID_N;
    const float* rp = rhs_p + (size_t)bp * GRID_N;
    float s = 0.0f;
    for (int g = 0; g < GRID_N; g++) s += A[(size_t)g * GRID_N + h] * rp[g];
    int tt = h >> 6, xx = h & 63;
    float bm = (tt == 0 || tt == ST - 1 || xx == 0 || xx == SX - 1) ? 1.0f : 0.0f;
    rhsv[i] = s + MU_ * bm * u_ch[i];
}

// ------------------------------------------------------------ blocked Cholesky
__global__ void potrf_diag(float* __restrict__ M, int o) {
    __shared__ float T[64 * 64];
    int tid = threadIdx.x;
    for (int idx = tid; idx < 4096; idx += blockDim.x)
        T[idx] = M[(size_t)(o + idx / 64) * GRID_N + o + (idx & 63)];
    __syncthreads();
    for (int c = 0; c < 64; c++) {
        if (tid == 0) T[c * 64 + c] = sqrtf(T[c * 64 + c]);
        __syncthreads();
        float d = T[c * 64 + c];
        for (int i = c + 1 + tid; i < 64; i += blockDim.x) T[i * 64 + c] /= d;
        __syncthreads();
        for (int idx = tid; idx < 4096; idx += blockDim.x) {
            int i = idx / 64, j = idx & 63;
            if (i > c && j > c && j <= i) T[i * 64 + j] -= T[i * 64 + c] * T[j * 64 + c];
        }
        __syncthreads();
    }
    for (int idx = tid; idx < 4096; idx += blockDim.x) {
        int i = idx / 64, j = idx & 63;
        M[(size_t)(o + i) * GRID_N + o + j] = (j <= i) ? T[idx] : 0.0f;
    }
}

__global__ void trsm_panel(float* __restrict__ M, int o) {
    __shared__ float Ld[64 * 64];
    int tid = threadIdx.x;
    for (int idx = tid; idx < 4096; idx += blockDim.x)
        Ld[idx] = M[(size_t)(o + idx / 64) * GRID_N + o + (idx & 63)];
    __syncthreads();
    int r = o + 64 + blockIdx.x * blockDim.x + tid;
    if (r >= GRID_N) return;
    float v[64];
    float* row = M + (size_t)r * GRID_N + o;
#pragma unroll 1
    for (int c = 0; c < 64; c++) {
        float s = row[c];
        for (int j = 0; j < c; j++) s -= v[j] * Ld[c * 64 + j];
        v[c] = s / Ld[c * 64 + c];
    }
    for (int c = 0; c < 64; c++) row[c] = v[c];
}

// trailing update: M[i,j] -= P_i . P_j^T, K=64, lower tiles only (WMMA f32)
__global__ void syrk_update(float* __restrict__ M, int o) {
    if (blockIdx.y > blockIdx.x) return;                  // uniform: whole wave exits
    int lane = threadIdx.x;
    int base = o + 64;
    int i0 = base + blockIdx.x * 16;
    int j0 = base + blockIdx.y * 16;
    int m = lane & 15, kh = lane >> 4;
    v8f c = {};
    for (int k0 = 0; k0 < 64; k0 += 4) {
        int k = k0 + 2 * kh;
        v2f a, b;
        a.x = M[(size_t)(i0 + m) * GRID_N + o + k];
        a.y = M[(size_t)(i0 + m) * GRID_N + o + k + 1];
        b.x = M[(size_t)(j0 + m) * GRID_N + o + k];
        b.y = M[(size_t)(j0 + m) * GRID_N + o + k + 1];
        c = __builtin_amdgcn_wmma_f32_16x16x4_f32(false, a, false, b,
                                                  (short)0, c, false, false);
    }
    for (int r = 0; r < 8; r++) {
        int i = i0 + r + 8 * kh;
        M[(size_t)i * GRID_N + j0 + m] -= c[r];
    }
}

// one wave per RHS; shuffle-reduced dot products; y kept in LDS
__global__ void chol_solve(const float* __restrict__ L, const float* __restrict__ rhsv,
                           float* __restrict__ u0) {
    __shared__ float yb[4 * GRID_N];
    int w = threadIdx.x >> 5, lane = threadIdx.x & 31;
    float* y = yb + w * GRID_N;
    const float* b = rhsv + (size_t)w * GRID_N;
    for (int i = 0; i < GRID_N; i++) {                    // forward: L y = b
        const float* Li = L + (size_t)i * GRID_N;
        float s = 0.0f;
        for (int j = lane; j < i; j += 32) s += Li[j] * y[j];
        for (int off = 16; off > 0; off >>= 1) s += __shfl_down(s, off, 32);
        if (lane == 0) y[i] = (b[i] - s) / Li[i];
        __syncthreads();
    }
    for (int i = GRID_N - 1; i >= 0; i--) {               // backward: L^T x = y
        float s = 0.0f;
        for (int j = i + 1 + lane; j < GRID_N; j += 32) s += L[(size_t)j * GRID_N + i] * y[j];
        for (int off = 16; off > 0; off >>= 1) s += __shfl_down(s, off, 32);
        if (lane == 0) {
            float xi = (y[i] - s) / L[(size_t)i * GRID_N + i];
            y[i] = xi;
            u0[(size_t)w * GRID_N + i] = xi;
        }
        __syncthreads();
    }
}

// ------------------------------------------- derivatives, residual max, output
__global__ void init_eps(float* eps_out) { *eps_out = 0.0f; }

__global__ void deriv_eps(const float* __restrict__ u0, const float* __restrict__ rhs_p,
                          const float* __restrict__ coef, float* __restrict__ up_j,
                          float* __restrict__ dt_j, float* __restrict__ dx_j,
                          float* __restrict__ dxx_j, float* __restrict__ eps_out) {
    int i = blockIdx.x * blockDim.x + threadIdx.x;
    if (i >= 4 * GRID_N) return;
    int bp = i / GRID_N, g = i % GRID_N;
    int tt = g >> 6, xx = g & 63;
    const float* u = u0 + (size_t)bp * GRID_N;
    float uc = u[g];
    float dt;
    if (tt == 0)            dt = (u[g + SX] - u[g]) / HT;
    else if (tt == ST - 1)  dt = (u[g] - u[g - SX]) / HT;
    else                    dt = (u[g + SX] - u[g - SX]) * (0.5f / HT);
    float dx;
    if (xx == 0)            dx = (u[g + 1] - u[g]) / HX;
    else if (xx == SX - 1)  dx = (u[g] - u[g - 1]) / HX;
    else                    dx = (u[g + 1] - u[g - 1]) * (0.5f / HX);
    int t0 = (tt == 0) ? 0 : (tt == ST - 1 ? ST - 3 : tt - 1);
    int gb = t0 * SX + xx;
    float dtt = (u[gb] - 2.0f * u[gb + SX] + u[gb + 2 * SX]) / (HT * HT);
    int x0 = (xx == 0) ? 0 : (xx == SX - 1 ? SX - 3 : xx - 1);
    gb = tt * SX + x0;
    float dxx = (u[gb] - 2.0f * u[gb + 1] + u[gb + 2]) / (HX * HX);
    int pt = bp >> 1, px = bp & 1;
    int idx = (pt * ST + tt) * X_FULL + px * SX + xx;      // join_patches
    up_j[idx] = uc; dt_j[idx] = dt; dx_j[idx] = dx; dxx_j[idx] = dxx;
    float Au = coef[0] * uc + coef[1] * dt + coef[2] * dx + coef[3] * dtt + coef[4] * dxx;
    float r = fabsf(Au - rhs_p[i]);
    atomicMax((unsigned int*)eps_out, __float_as_uint(r)); // valid: r >= 0
}

// ---------------------------------------------------- fused MLP (1 WG per net)
__global__ void mlp_kernel(const float* w1a, const float* b1a, const float* w2a, const float* b2a,
                           const float* w3a, const float* b3a,
                           const float* w1b, const float* b1b, const float* w2b, const float* b2b,
                           const float* w3b, const float* b3b,
                           const float* pin1, const float* pin2, float* __restrict__ params_out) {
    __shared__ float pin[256];
    __shared__ float h1[1024];
    __shared__ float h2[1024];
    int net = blockIdx.x, tid = threadIdx.x;
    const float* W1 = net ? w1b : w1a; const float* B1 = net ? b1b : b1a;
    const float* W2 = net ? w2b : w2a; const float* B2 = net ? b2b : b2a;
    const float* W3 = net ? w3b : w3a; const float* B3 = net ? b3b : b3a;
    const float* P  = net ? pin2 : pin1;
    if (tid < 256) pin[tid] = P[tid];
    __syncthreads();
    float s = B1[tid];
    for (int j = 0; j < 256; j++) s += W1[(size_t)tid * 256 + j] * pin[j];
    h1[tid] = fmaxf(s, 0.0f);
    __syncthreads();
    s = B2[tid];
    for (int j = 0; j < 1024; j++) s += W2[(size_t)tid * 1024 + j] * h1[j];
    h2[tid] = fmaxf(s, 0.0f);
    __syncthreads();
    if (tid < 3) {
        float o = B3[tid];
        for (int j = 0; j < 1024; j++) o += W3[(size_t)tid * 1024 + j] * h2[j];
        params_out[net * 3 + tid] = o;
    }
}

__global__ void final_eq(const float* __restrict__ up_j, const float* __restrict__ dt_j,
                         const float* __restrict__ dx_j, const float* __restrict__ dxx_j,
                         const float* __restrict__ params, float* __restrict__ out) {
    int i = blockIdx.x * blockDim.x + threadIdx.x;
    if (i >= NPIX) return;
    float up = up_j[i];
    float p = params[0] + params[1] * up + params[2] * up * up;
    float q = params[3] + params[4] * up + params[5] * up * up;
    out[i] = up;
    out[NPIX + i] = dt_j[i] - p * dx_j[i] - q * dxx_j[i];
}

// ------------------------------------------------------------------- launcher
extern "C" void kernel_launch(void* const* d_in, const int* in_sizes, int n_in,
                              void* d_out, int out_size, void* d_ws, size_t ws_size,
                              hipStream_t stream) {
    const float* u = (const float*)d_in[0];
    const float* cw[7]; const float* cb[7];
    for (int i = 0; i < 7; i++) { cw[i] = (const float*)d_in[3 + 2 * i]; cb[i] = (const float*)d_in[4 + 2 * i]; }
    const float* p1w1 = (const float*)d_in[17]; const float* p1b1 = (const float*)d_in[18];
    const float* p1w2 = (const float*)d_in[19]; const float* p1b2 = (const float*)d_in[20];
    const float* p1w3 = (const float*)d_in[21]; const float* p1b3 = (const float*)d_in[22];
    const float* p2w1 = (const float*)d_in[23]; const float* p2b1 = (const float*)d_in[24];
    const float* p2w2 = (const float*)d_in[25]; const float* p2b2 = (const float*)d_in[26];
    const float* p2w3 = (const float*)d_in[27]; const float* p2b3 = (const float*)d_in[28];
    const float* pin1 = (const float*)d_in[29];
    const float* pin2 = (const float*)d_in[30];
    const float* coef = (const float*)d_in[31];

    float* ws    = (float*)d_ws;
    float* h0    = ws;                     // 64*7680
    float* h1    = ws + 491520;            // 64*7680
    float* rhsf  = ws + 983040;            // 7680
    float* A     = ws + 990720;            // 1920*1920
    float* Mm    = ws + 4677120;           // 1920*1920
    float* u_ch  = ws + 8363520;           // 4*1920
    float* rhs_p = ws + 8371200;
    float* rhsv  = ws + 8378880;
    float* u0    = ws + 8386560;
    float* up_j  = ws + 8394240;           // 7680 each
    float* dt_j  = ws + 8401920;
    float* dx_j  = ws + 8409600;
    float* dxx_j = ws + 8417280;
    float* wtb   = ws + 8424960;           // 25*64*64 transposed conv weights
    float* out   = (float*)d_out;

    // conv stack
    conv_first<<<(64 * NPIX + 255) / 256, 256, 0, stream>>>(u, cw[0], cb[0], h0);
    float* bufs[2] = { h0, h1 };
    for (int l = 1; l <= 5; l++) {
        transpose_w<<<(25 * 4096 + 255) / 256, 256, 0, stream>>>(cw[l], wtb);
        conv_mid_wmma<<<480, 128, 0, stream>>>(bufs[(l - 1) & 1], wtb, cb[l], bufs[l & 1]);
    }
    conv_last<<<(NPIX + 255) / 256, 256, 0, stream>>>(h1, cw[6], cb[6], rhsf);

    // operator + normal equations
    build_A<<<GRID_N, 256, 0, stream>>>(A, coef);
    gemm_ata<<<dim3(60, 60), 128, 0, stream>>>(A, Mm);
    add_reg<<<(GRID_N + 255) / 256, 256, 0, stream>>>(Mm);
    gather_patches<<<(4 * GRID_N + 255) / 256, 256, 0, stream>>>(u, rhsf, u_ch, rhs_p);
    build_rhsv<<<(4 * GRID_N + 255) / 256, 256, 0, stream>>>(A, rhs_p, u_ch, rhsv);

    // blocked Cholesky, NB=64
    for (int kp = 0; kp < 30; kp++) {
        int o = kp * 64;
        potrf_diag<<<1, 256, 0, stream>>>(Mm, o);
        int rows = GRID_N - o - 64;
        if (rows > 0) {
            trsm_panel<<<(rows + 127) / 128, 128, 0, stream>>>(Mm, o);
            int nt = rows / 16;
            syrk_update<<<dim3(nt, nt), 32, 0, stream>>>(Mm, o);
        }
    }
    chol_solve<<<1, 128, 0, stream>>>(Mm, rhsv, u0);

    // derivatives, residual max, MLP params, PDE residual
    init_eps<<<1, 1, 0, stream>>>(out + 15360);
    deriv_eps<<<(4 * GRID_N + 255) / 256, 256, 0, stream>>>(u0, rhs_p, coef,
                                                            up_j, dt_j, dx_j, dxx_j, out + 15360);
    mlp_kernel<<<2, 1024, 0, stream>>>(p1w1, p1b1, p1w2, p1b2, p1w3, p1b3,
                                       p2w1, p2b1, p2w2, p2b2, p2w3, p2b3,
                                       pin1, pin2, out + 15361);
    final_eq<<<(NPIX + 255) / 256, 256, 0, stream>>>(up_j, dt_j, dx_j, dxx_j, out + 15361, out);
}